// FeatureReweightingModule_72447508349057
// MI455X (gfx1250) — compile-verified
//
#include <hip/hip_runtime.h>
#include <hip/hip_bf16.h>
#include <stdint.h>

// FeatureReweightingModule: out[b,n,c,s] = q[b,s,c] * w[n,c]
// B=8, S=2048, D=256, N=20. f32 in / f32 out.
// Pure broadcast-scale -> HBM-write-bound (~352 MB traffic, ~15us floor @23.3TB/s).
// Strategy: async-DMA q tiles into LDS *transposed* (global reads coalesced in c,
// per-lane LDS scatter to [c][s]); then 128-bit coalesced stores along s, reusing
// each q element for all 20 classes. Async addresses are strength-reduced to
// base + constant stride (4096B global / 16B LDS per step) to minimize VALU.

typedef __attribute__((ext_vector_type(4))) float v4f;

#define B_  8
#define S_  2048
#define D_  256
#define N_  20
#define TS  64          // s-tile
#define TC  64          // c-tile
#define LSTRIDE 68      // padded LDS row stride (floats); b128 reads bank-conflict-free

__device__ __forceinline__ void async_load_f32_to_lds(uint32_t lds_byte_off,
                                                      const float* gaddr) {
  // GLOBAL_LOAD_ASYNC_TO_LDS_B32, GV mode (64-bit VGPR address, saddr=off).
  // VDST = per-lane LDS byte address, tracked by ASYNCcnt.
  asm volatile("global_load_async_to_lds_b32 %0, %1, off"
               :
               : "v"(lds_byte_off), "v"((unsigned long long)(uintptr_t)gaddr)
               : "memory");
}

__global__ __launch_bounds__(256) void frw_scale_kernel(const float* __restrict__ q,
                                                        const float* __restrict__ w,
                                                        float* __restrict__ out) {
  __shared__ __align__(16) float tile[TC * LSTRIDE];  // q tile, [c][s] (transposed), padded
  __shared__ float wtile[N_ * TC];                    // weight slice, [n][c]

  const int tid = threadIdx.x;
  const int bid = blockIdx.x;          // 0..1023
  const int b   = bid >> 7;            // 128 tiles per batch image
  const int r   = bid & 127;
  const int s0  = (r >> 2) * TS;       // 32 s-tiles
  const int c0  = (r & 3) * TC;        // 4 c-tiles

  // LDS byte offsets (low 32 bits of the generic address == LDS offset)
  const uint32_t tile_off = (uint32_t)(uintptr_t)&tile[0];
  const uint32_t w_off    = (uint32_t)(uintptr_t)&wtile[0];

  // ---- Async DMA: q tile, transposed into LDS ----
  // Step = 256 threads = multiple of TC, so per-lane c_local is loop-invariant:
  // global advances 4 s-rows (4096 B), LDS advances 4 floats (16 B) per step.
  {
    const int cl  = tid & (TC - 1);                  // fast: c (coalesced global)
    const int sl0 = tid >> 6;                        // starting s row (0..3)
    const float* g = q + (size_t)(b * S_ + s0 + sl0) * D_ + (c0 + cl);
    uint32_t lo = tile_off + (uint32_t)((cl * LSTRIDE + sl0) * 4);
#pragma unroll
    for (int it = 0; it < (TS * TC) / 256; ++it) {   // 16 async b32 per lane
      async_load_f32_to_lds(lo, g);
      g  += 4 * D_;                                  // +4096 B
      lo += 4 * 4;                                   // +16 B along s
    }
  }
  // ---- Async DMA: weight slice w[0..19, c0..c0+63] (same invariance trick) ----
  {
    const int cc = tid & 63;
    const int n0 = tid >> 6;                         // starting n row (0..3)
    const float* gw = w + n0 * D_ + c0 + cc;
    uint32_t wo = w_off + (uint32_t)((n0 * TC + cc) * 4);
#pragma unroll
    for (int it = 0; it < (N_ * TC) / 256; ++it) {   // 5 async b32 per lane
      async_load_f32_to_lds(wo, gw);
      gw += 4 * D_;                                  // +4 n-rows in global
      wo += 4 * TC * 4;                              // +1024 B in LDS
    }
  }

  asm volatile("s_wait_asynccnt 0x0" ::: "memory");  // my wave's DMAs landed in LDS
  __syncthreads();                                   // everyone's landed

  // ---- Compute + store: 128-bit stores along s, q reused for all 20 classes ----
  const int ss  = (tid & 15) * 4;   // 16 lanes x float4 = 64 s (256B burst / half-wave)
  const int ccb = tid >> 4;         // 16 c rows per pass

#pragma unroll
  for (int ci = 0; ci < 4; ++ci) {
    const int cc = ccb + ci * 16;
    const v4f v = *(const v4f*)&tile[cc * LSTRIDE + ss];  // conflict-free b128
    float* op = out + ((size_t)(b * N_) * D_ + (size_t)(c0 + cc)) * S_ + (s0 + ss);
#pragma unroll
    for (int n = 0; n < N_; ++n) {
      const float wv = wtile[n * TC + cc];  // 16-lane-uniform addr -> LDS broadcast
      *(v4f*)op = v * wv;
      op += (size_t)D_ * S_;                // next class plane
    }
  }
}

extern "C" void kernel_launch(void* const* d_in, const int* in_sizes, int n_in,
                              void* d_out, int out_size, void* d_ws, size_t ws_size,
                              hipStream_t stream) {
  (void)in_sizes; (void)n_in; (void)out_size; (void)d_ws; (void)ws_size;
  const float* q = (const float*)d_in[0];   // (8, 2048, 256)
  const float* w = (const float*)d_in[1];   // (20, 256)
  float* out = (float*)d_out;               // (160, 256, 2048)

  frw_scale_kernel<<<dim3(B_ * (S_ / TS) * (D_ / TC)), dim3(256), 0, stream>>>(q, w, out);
}